// QuantumBornMachine_9010841387850
// MI455X (gfx1250) — compile-verified
//
#include <hip/hip_runtime.h>
#include <hip/hip_bf16.h>
#include <math.h>

#define NQ 20
#define NSTATE (1u << NQ)   // 1048576

typedef __attribute__((ext_vector_type(2))) float v2f;
typedef __attribute__((ext_vector_type(8))) float v8f;

struct cplx { float re, im; };
__device__ __forceinline__ cplx cmul(cplx a, cplx b) {
    return { a.re * b.re - a.im * b.im, a.re * b.im + a.im * b.re };
}

// Inverse of the CNOT-ring basis permutation (qubit i lives at bit 19-i):
//   x_b = y_b ^ y_{b+1} for b<=17,  x_19 = y_19 ^ y_0,  x_18 = y_18 ^ y_19 ^ y_0
__device__ __forceinline__ unsigned pinv(unsigned y) {
    unsigned x   = (y ^ (y >> 1)) & 0x3FFFFu;
    unsigned b19 = ((y >> 19) ^ y) & 1u;
    unsigned b18 = ((y >> 18) & 1u) ^ b19;
    return x | (b19 << 19) | (b18 << 18);
}

// CZ pairs (i,i+2), i=0,2,..,16 -> parity of popc(y & (y>>2) & 0x2AAAA)
__device__ __forceinline__ float czsign(unsigned y) {
    return (__popc(y & (y >> 2) & 0x2AAAAu) & 1) ? -1.0f : 1.0f;
}

// ---------------------------------------------------------------------------
// Gate prep: for each (layer l, group g of 4 qubits) build the 16x16 complex
// unitary Kron(U_{4g},U_{4g+1},U_{4g+2},U_{4g+3}) with U_q = Rz*Ry*Rx fused.
// Stores Ur, Ui, and -Ui (f32 WMMA has no A-negate): 3*256 floats per block.
// Grid: 15 blocks (l*5+g), 256 threads (one (i,j) entry each).
// ---------------------------------------------------------------------------
__global__ void qbm_prep_gates(const float* __restrict__ theta,
                               float* __restrict__ mats) {
    int blk = blockIdx.x;            // l*5 + g
    int l = blk / 5, g = blk % 5;
    int t = threadIdx.x;             // 0..255
    int i = t >> 4, j = t & 15;

    cplx prod = {1.0f, 0.0f};
    #pragma unroll
    for (int m = 0; m < 4; ++m) {
        int q = 4 * g + m;
        int p = l * 60 + q * 3;
        float hx = 0.5f * theta[p + 0];
        float hy = 0.5f * theta[p + 1];
        float hz = 0.5f * theta[p + 2];
        float sx, cx, sy, cy, sz, cz;
        sincosf(hx, &sx, &cx);
        sincosf(hy, &sy, &cy);
        sincosf(hz, &sz, &cz);
        // Ry*Rx:
        //  (0,0)= cy*cx + i sy*sx   (0,1)= -sy*cx - i cy*sx
        //  (1,0)= sy*cx - i cy*sx   (1,1)=  cy*cx - i sy*sx
        int bi = (i >> (3 - m)) & 1;
        int bj = (j >> (3 - m)) & 1;
        cplx e;
        if (bi == 0) {
            e = bj == 0 ? cplx{ cy * cx,  sy * sx } : cplx{ -sy * cx, -cy * sx };
        } else {
            e = bj == 0 ? cplx{ sy * cx, -cy * sx } : cplx{  cy * cx, -sy * sx };
        }
        // Rz row phase: e^{-i hz} for row 0, e^{+i hz} for row 1
        cplx ph = { cz, bi ? sz : -sz };
        e = cmul(ph, e);
        prod = cmul(prod, e);
    }
    float* base = mats + (size_t)blk * 768;
    base[t]       =  prod.re;   // Ur
    base[256 + t] =  prod.im;   // Ui
    base[512 + t] = -prod.im;   // -Ui
}

// ---------------------------------------------------------------------------
// Apply a 16x16 complex unitary to 4 qubits whose bits occupy the contiguous
// index field [s, s+3].  State reshapes as (H, 16, L=2^s); for each of the
// 2^16 (h,low) rows:  new[j] = sum_k U[j,k] * old[k].
// GEMM via V_WMMA_F32_16X16X4_F32:  D(16j x 16r) = A(U, 16x4) * B(S^T, 4x16),
// chained over 4 K-chunks; complex = 4 real GEMMs (16 WMMAs / wave-tile):
//   D_re += Ur*Sr ; D_re += (-Ui)*Si ; D_im += Ur*Si ; D_im += Ui*Sr
//
// MODE 0: in-place (src==dst); each wave reads exactly the rows it writes.
// MODE 1: fused CNOT-ring(+CZ if DO_CZ) permutation from the PREVIOUS layer:
//         load sgn(e)*src[pinv(e)], write dst[e] (src != dst, bijective).
// MODE 2: constant uniform-superposition input (Re=2^-10, Im=0); no loads.
// Compile-time MODE/DO_CZ -> straight-line code, clause-packed loads.
// Grid: 512 blocks x 256 threads = 4096 waves, 16 rows each.
// ---------------------------------------------------------------------------
template <int MODE, int DO_CZ>
__global__ void qbm_apply4(const float* __restrict__ srcRe, const float* __restrict__ srcIm,
                           float* __restrict__ dstRe, float* __restrict__ dstIm,
                           const float* __restrict__ mats, int matBlk, int s) {
    const float* Ur = mats + (size_t)matBlk * 768;
    const float* Ui = Ur + 256;
    const float* Un = Ur + 512;

    unsigned lane = threadIdx.x & 31u;
    unsigned wave = blockIdx.x * (blockDim.x >> 5) + (threadIdx.x >> 5);
    unsigned half = lane >> 4;       // 0 or 1
    unsigned n    = lane & 15u;      // row within tile / matrix row M

    unsigned r    = wave * 16u + n;  // global row id in [0, 2^16)
    unsigned L    = 1u << s;
    unsigned h    = r >> s;
    unsigned low  = r & (L - 1u);
    unsigned rowbase = (h << (s + 4)) + low;   // element index of field value 0

    v8f accRe = {};
    v8f accIm = {};

    #pragma unroll
    for (int c = 0; c < 4; ++c) {
        int ka = 4 * c + 2 * (int)half;    // A/B layout: VGPR0 holds K=ka,
        int kb = ka + 1;                   //             VGPR1 holds K=ka+1
        v2f ar, ai, an, bre, bim;
        ar.x = Ur[n * 16 + ka];  ar.y = Ur[n * 16 + kb];
        ai.x = Ui[n * 16 + ka];  ai.y = Ui[n * 16 + kb];
        an.x = Un[n * 16 + ka];  an.y = Un[n * 16 + kb];

        unsigned ea = rowbase + ((unsigned)ka << s);
        unsigned eb = rowbase + ((unsigned)kb << s);

        if (MODE == 2) {
            bre.x = 0.0009765625f; bre.y = 0.0009765625f;   // 2^-10
            bim.x = 0.0f;          bim.y = 0.0f;
        } else if (MODE == 1) {
            unsigned la = pinv(ea), lb = pinv(eb);
            float sa = DO_CZ ? czsign(ea) : 1.0f;
            float sb = DO_CZ ? czsign(eb) : 1.0f;
            bre.x = sa * srcRe[la];  bre.y = sb * srcRe[lb];
            bim.x = sa * srcIm[la];  bim.y = sb * srcIm[lb];
        } else {
            bre.x = srcRe[ea];  bre.y = srcRe[eb];
            bim.x = srcIm[ea];  bim.y = srcIm[eb];
        }

        accRe = __builtin_amdgcn_wmma_f32_16x16x4_f32(false, ar, false, bre,
                                                      (short)0, accRe, false, false);
        accRe = __builtin_amdgcn_wmma_f32_16x16x4_f32(false, an, false, bim,
                                                      (short)0, accRe, false, false);
        accIm = __builtin_amdgcn_wmma_f32_16x16x4_f32(false, ar, false, bim,
                                                      (short)0, accIm, false, false);
        accIm = __builtin_amdgcn_wmma_f32_16x16x4_f32(false, ai, false, bre,
                                                      (short)0, accIm, false, false);
    }

    // D layout: lane holds column r = n; VGPR v holds row j = v + 8*half.
    #pragma unroll
    for (int v = 0; v < 8; ++v) {
        unsigned j = (unsigned)v + 8u * half;
        unsigned e = rowbase + (j << s);
        dstRe[e] = accRe[v];
        dstIm[e] = accIm[v];
    }
}

// ---------------------------------------------------------------------------
// Born probabilities fused with the LAST layer's CNOT-ring permutation.
// CZ sign of the last layer is irrelevant under |.|^2.
// ---------------------------------------------------------------------------
__global__ void qbm_probs_perm(const float* __restrict__ re, const float* __restrict__ im,
                               float* __restrict__ out) {
    unsigned y = blockIdx.x * blockDim.x + threadIdx.x;
    unsigned x = pinv(y);
    float r = re[x], m = im[x];
    out[y] = r * r + m * m;
}

// ---------------------------------------------------------------------------
extern "C" void kernel_launch(void* const* d_in, const int* in_sizes, int n_in,
                              void* d_out, int out_size, void* d_ws, size_t ws_size,
                              hipStream_t stream) {
    (void)in_sizes; (void)n_in; (void)out_size; (void)ws_size;
    const float* theta = (const float*)d_in[0];      // 180 f32
    float* out = (float*)d_out;                      // 2^20 f32
    float* ws  = (float*)d_ws;

    const size_t N = NSTATE;
    float* Re0  = ws;
    float* Im0  = ws + N;
    float* Re1  = ws + 2 * N;
    float* Im1  = ws + 3 * N;
    float* mats = ws + 4 * N;                        // 15 * 768 f32

    qbm_prep_gates<<<15, 256, 0, stream>>>(theta, mats);

    // ---- Layer 0: g0 consumes the uniform superposition as a constant -----
    qbm_apply4<2, 0><<<512, 256, 0, stream>>>(Re0, Im0, Re0, Im0, mats, 0, 16);
    for (int g = 1; g < 5; ++g)
        qbm_apply4<0, 0><<<512, 256, 0, stream>>>(Re0, Im0, Re0, Im0, mats,
                                                  g, 16 - 4 * g);

    // ---- Layer 1: g0 fuses layer-0 tail (CNOT ring + CZ), buf0 -> buf1 ----
    qbm_apply4<1, 1><<<512, 256, 0, stream>>>(Re0, Im0, Re1, Im1, mats, 5, 16);
    for (int g = 1; g < 5; ++g)
        qbm_apply4<0, 0><<<512, 256, 0, stream>>>(Re1, Im1, Re1, Im1, mats,
                                                  5 + g, 16 - 4 * g);

    // ---- Layer 2: g0 fuses layer-1 tail (CNOT ring, no CZ), buf1 -> buf0 --
    qbm_apply4<1, 0><<<512, 256, 0, stream>>>(Re1, Im1, Re0, Im0, mats, 10, 16);
    for (int g = 1; g < 5; ++g)
        qbm_apply4<0, 0><<<512, 256, 0, stream>>>(Re0, Im0, Re0, Im0, mats,
                                                  10 + g, 16 - 4 * g);

    // ---- Probabilities fused with layer-2 tail permutation ----------------
    qbm_probs_perm<<<NSTATE / 256, 256, 0, stream>>>(Re0, Im0, out);
}